// FeatModel_50611894616411
// MI455X (gfx1250) — compile-verified
//
#include <hip/hip_runtime.h>
#include <math.h>

// ---------------- problem constants ----------------
#define BB   8
#define PP   196
#define DD   768
#define HH_  12
#define DHH  64
#define LL   4
#define NL   1568              // left nodes  (8*196)
#define NT   3136              // total nodes (2*NL)
#define THRESH  0.1f
#define SLOPE   0.2f
#define LOG2F_  0.6931471805599453f

typedef __attribute__((ext_vector_type(2))) float v2f;
typedef __attribute__((ext_vector_type(8))) float v8f;

#define WMMA_F32(a, b, c) \
    __builtin_amdgcn_wmma_f32_16x16x4_f32(false, (a), false, (b), (short)0, (c), false, false)

__device__ __forceinline__ float leaky(float x) { return x >= 0.f ? x : SLOPE * x; }

// ---------------- row L2-normalize (one block per row) ----------------
__global__ void k_normalize(const float* __restrict__ src, float* __restrict__ dst) {
    __shared__ float red[256];
    int row = blockIdx.x;
    const float* p = src + (size_t)row * DD;
    float ss = 0.f;
    for (int j = threadIdx.x; j < DD; j += 256) { float v = p[j]; ss += v * v; }
    red[threadIdx.x] = ss; __syncthreads();
    for (int s = 128; s > 0; s >>= 1) {
        if (threadIdx.x < s) red[threadIdx.x] += red[threadIdx.x + s];
        __syncthreads();
    }
    float inv = rsqrtf(red[0]);
    float* q = dst + (size_t)row * DD;
    for (int j = threadIdx.x; j < DD; j += 256) q[j] = p[j] * inv;
}

// ---------------- build h = concat(node_l, node_r) ----------------
__global__ void k_build_h(const float* __restrict__ lf, const float* __restrict__ rf,
                          float* __restrict__ h) {
    size_t idx = (size_t)blockIdx.x * blockDim.x + threadIdx.x;
    size_t half = (size_t)NL * DD;
    if (idx < half)            h[idx] = lf[idx];
    else if (idx < 2 * half)   h[idx] = rf[idx - half];
}

// ---------------- mask = (nr @ nl^T > THRESH) via f32 WMMA ----------------
// one wave computes a 16x32 tile: A fragment reused across 2 column tiles;
// B = nl^T so both fragments are contiguous float2 loads
__global__ void k_mask_wmma(const float* __restrict__ nr_, const float* __restrict__ nl_,
                            unsigned char* __restrict__ mask) {
    int lane = threadIdx.x;
    int rloc = lane & 15, hi = lane >> 4;
    int r0 = blockIdx.y * 16, n0 = blockIdx.x * 32;
    const float* arow  = nr_ + (size_t)(r0 + rloc) * DD;        // A row (M = r)
    const float* brow0 = nl_ + (size_t)(n0 + rloc) * DD;        // B cols n0..n0+15
    const float* brow1 = nl_ + (size_t)(n0 + 16 + rloc) * DD;   // B cols n0+16..n0+31
    v8f c0 = {}, c1 = {};
    for (int k = 0; k < DD; k += 4) {
        if ((k & 63) == 0) {
            __builtin_prefetch(arow + k + 64, 0, 1);
            __builtin_prefetch(brow0 + k + 64, 0, 1);
            __builtin_prefetch(brow1 + k + 64, 0, 1);
        }
        int kk = k + 2 * hi;
        v2f a  = *(const v2f*)(arow + kk);
        v2f b0 = *(const v2f*)(brow0 + kk);
        v2f b1 = *(const v2f*)(brow1 + kk);
        c0 = WMMA_F32(a, b0, c0);
        c1 = WMMA_F32(a, b1, c1);
    }
    for (int v = 0; v < 8; ++v) {
        size_t rr = (size_t)(r0 + 8 * hi + v) * NL;
        mask[rr + n0 + rloc]      = (c0[v] > THRESH) ? 1 : 0;
        mask[rr + n0 + 16 + rloc] = (c1[v] > THRESH) ? 1 : 0;
    }
}

// ---------------- C = A(MxK) @ B(KxN), row-major, f32 WMMA ----------------
// one wave computes a 16x64 strip: one A fragment feeds 4 WMMAs per K-step
__global__ void k_gemm_wmma(const float* __restrict__ A, const float* __restrict__ Bm,
                            float* __restrict__ C, int M, int N, int K) {
    int lane = threadIdx.x;
    int rloc = lane & 15, hi = lane >> 4;
    int m0 = blockIdx.y * 16, n0 = blockIdx.x * 64;
    const float* arow = A + (size_t)(m0 + rloc) * K;
    const float* bcol = Bm + (size_t)(n0 + rloc);
    v8f c0 = {}, c1 = {}, c2 = {}, c3 = {};
    for (int k = 0; k < K; k += 4) {
        if ((k & 63) == 0) __builtin_prefetch(arow + k + 64, 0, 1);
        int kk = k + 2 * hi;
        v2f a = *(const v2f*)(arow + kk);
        const float* b0 = bcol + (size_t)kk * N;
        const float* b1 = b0 + N;
        v2f b;
        b.x = b0[0];  b.y = b1[0];
        c0 = WMMA_F32(a, b, c0);
        b.x = b0[16]; b.y = b1[16];
        c1 = WMMA_F32(a, b, c1);
        b.x = b0[32]; b.y = b1[32];
        c2 = WMMA_F32(a, b, c2);
        b.x = b0[48]; b.y = b1[48];
        c3 = WMMA_F32(a, b, c3);
    }
    for (int v = 0; v < 8; ++v) {
        float* crow = C + (size_t)(m0 + 8 * hi + v) * N + n0 + rloc;
        crow[0]  = c0[v];
        crow[16] = c1[v];
        crow[32] = c2[v];
        crow[48] = c3[v];
    }
}

// ---------------- el/er: per-(node,head) 64-dots ----------------
__global__ void k_elr(const float* __restrict__ z, const float* __restrict__ as,
                      const float* __restrict__ ad, float* __restrict__ el,
                      float* __restrict__ er) {
    int idx = blockIdx.x * blockDim.x + threadIdx.x;
    if (idx >= NT * HH_) return;
    int n = idx / HH_, hh = idx % HH_;
    const float* zp = z + (size_t)n * DD + hh * DHH;
    const float* sp = as + hh * DHH;
    const float* dp = ad + hh * DHH;
    float e1 = 0.f, e2 = 0.f;
    for (int d = 0; d < DHH; ++d) { float v = zp[d]; e1 += v * sp[d]; e2 += v * dp[d]; }
    el[idx] = e1; er[idx] = e2;
}

// ---------------- attention pass 1: online max/sum per (r,head) ----------------
__global__ void k_att_stats(const float* __restrict__ el, const float* __restrict__ er,
                            const unsigned char* __restrict__ mask,
                            float* __restrict__ mst, float* __restrict__ sst) {
    int wid = blockIdx.x * 8 + (threadIdx.x >> 5);
    int lane = threadIdx.x & 31;
    int r = wid / HH_, hh = wid % HH_;
    float err = er[(size_t)(NL + r) * HH_ + hh];
    const unsigned char* mrow = mask + (size_t)r * NL;
    float m = -1e30f, s = 0.f;
    for (int l = lane; l < NL; l += 32) {
        if (mrow[l]) {
            float lg = leaky(el[(size_t)l * HH_ + hh] + err);
            if (lg > m) { s = s * __expf(m - lg) + 1.f; m = lg; }
            else        { s += __expf(lg - m); }
        }
    }
    for (int off = 16; off > 0; off >>= 1) {
        float mo = __shfl_xor(m, off, 32);
        float so = __shfl_xor(s, off, 32);
        float nm = fmaxf(m, mo);
        s = s * __expf(m - nm) + so * __expf(mo - nm);
        m = nm;
    }
    if (lane == 0) {
        float lg = leaky(el[(size_t)(NL + r) * HH_ + hh] + err);   // self logit
        float nm = fmaxf(m, lg);
        s = s * __expf(m - nm) + __expf(lg - nm);
        mst[wid] = nm; sst[wid] = s;
    }
}

// ---------------- attention pass 2: out_r = alpha @ zl (+ self) via f32 WMMA ----
// one wave: 16 r-rows x one head x d=64 (4 column tiles); alpha tile built on the fly
__global__ void k_att_aggregate(const float* __restrict__ z, const float* __restrict__ el,
                                const float* __restrict__ er, const unsigned char* __restrict__ mask,
                                const float* __restrict__ mst, const float* __restrict__ sst,
                                const float* __restrict__ bias, float* __restrict__ hn) {
    int lane = threadIdx.x;
    int rloc = lane & 15, hi = lane >> 4;
    int rt = blockIdx.x * 16;
    int hh = blockIdx.y;
    int colbase = hh * DHH;

    int rg = rt + rloc;                       // r row this lane feeds into A-frag
    float err = er[(size_t)(NL + rg) * HH_ + hh];
    float mr  = mst[(size_t)rg * HH_ + hh];
    float isr = 1.f / sst[(size_t)rg * HH_ + hh];
    const unsigned char* mrow = mask + (size_t)rg * NL;

    v8f c0 = {}, c1 = {}, c2 = {}, c3 = {};
    for (int k = 0; k < NL; k += 4) {
        int kk = k + 2 * hi;
        v2f a;
        {
            float l0 = leaky(el[(size_t)kk * HH_ + hh] + err);
            a.x = mrow[kk] ? __expf(l0 - mr) * isr : 0.f;
            float l1 = leaky(el[(size_t)(kk + 1) * HH_ + hh] + err);
            a.y = mrow[kk + 1] ? __expf(l1 - mr) * isr : 0.f;
        }
        const float* b0 = z + (size_t)kk * DD + colbase + rloc;       // zl row kk
        const float* b1 = b0 + DD;                                    // zl row kk+1
        v2f b;
        b.x = b0[0];  b.y = b1[0];
        c0 = WMMA_F32(a, b, c0);
        b.x = b0[16]; b.y = b1[16];
        c1 = WMMA_F32(a, b, c1);
        b.x = b0[32]; b.y = b1[32];
        c2 = WMMA_F32(a, b, c2);
        b.x = b0[48]; b.y = b1[48];
        c3 = WMMA_F32(a, b, c3);
    }
    // epilogue: self-attention term + bias, store into right half of hn
    for (int v = 0; v < 8; ++v) {
        int rr = rt + 8 * hi + v;
        size_t nid = (size_t)(NL + rr);
        float lg = leaky(el[nid * HH_ + hh] + er[nid * HH_ + hh]);
        float f = __expf(lg - mst[(size_t)rr * HH_ + hh]) / sst[(size_t)rr * HH_ + hh];
        const float* zr = z + nid * DD + colbase + rloc;
        const float* bp = bias + colbase + rloc;
        float* op = hn + nid * DD + colbase + rloc;
        op[0]  = c0[v] + f * zr[0]  + bp[0];
        op[16] = c1[v] + f * zr[16] + bp[16];
        op[32] = c2[v] + f * zr[32] + bp[32];
        op[48] = c3[v] + f * zr[48] + bp[48];
    }
}

// ---------------- hn[0:rows] = z + bias ----------------
__global__ void k_addbias(const float* __restrict__ z, const float* __restrict__ bias,
                          float* __restrict__ hn, int rows) {
    size_t idx = (size_t)blockIdx.x * blockDim.x + threadIdx.x;
    if (idx < (size_t)rows * DD) hn[idx] = z[idx] + bias[idx % DD];
}

// ---------------- in-place ELU ----------------
__global__ void k_elu(float* __restrict__ x, int n) {
    size_t idx = (size_t)blockIdx.x * blockDim.x + threadIdx.x;
    if (idx < (size_t)n) { float v = x[idx]; x[idx] = v > 0.f ? v : __expf(v) - 1.f; }
}

// ---------------- pool layer: softmax-pool each group into row p=0 ----------------
__global__ void k_pool(const float* __restrict__ z, const float* __restrict__ el,
                       const float* __restrict__ er, const float* __restrict__ bias,
                       float* __restrict__ hn) {
    int wid = blockIdx.x * 8 + (threadIdx.x >> 5);
    int lane = threadIdx.x & 31;
    if (wid >= (NT / PP) * HH_) return;
    int g = wid / HH_, hh = wid % HH_;
    int base = g * PP;
    float er0 = er[(size_t)base * HH_ + hh];
    float m = -1e30f, s = 0.f;
    for (int p = lane; p < PP; p += 32) {
        float lg = leaky(el[(size_t)(base + p) * HH_ + hh] + er0) + (p == 0 ? LOG2F_ : 0.f);
        if (lg > m) { s = s * __expf(m - lg) + 1.f; m = lg; }
        else        { s += __expf(lg - m); }
    }
    for (int off = 16; off > 0; off >>= 1) {
        float mo = __shfl_xor(m, off, 32);
        float so = __shfl_xor(s, off, 32);
        float nm = fmaxf(m, mo);
        s = s * __expf(m - nm) + so * __expf(mo - nm);
        m = nm;
    }
    float isr = 1.f / s;
    int d0 = 2 * lane;                                    // two d-lanes each
    float a0 = 0.f, a1 = 0.f;
    for (int p = 0; p < PP; ++p) {
        float lg = leaky(el[(size_t)(base + p) * HH_ + hh] + er0) + (p == 0 ? LOG2F_ : 0.f);
        float w = __expf(lg - m) * isr;
        const float* zp = z + (size_t)(base + p) * DD + hh * DHH + d0;
        a0 += w * zp[0]; a1 += w * zp[1];
    }
    float* o = hn + (size_t)base * DD + hh * DHH + d0;
    o[0] = a0 + bias[hh * DHH + d0];
    o[1] = a1 + bias[hh * DHH + d0 + 1];
}

// ---------------- gather h[::P] rows into output ----------------
__global__ void k_gather(const float* __restrict__ h, float* __restrict__ out) {
    int idx = blockIdx.x * blockDim.x + threadIdx.x;
    if (idx >= (NT / PP) * DD) return;
    int g = idx / DD, j = idx % DD;
    out[idx] = h[(size_t)g * PP * DD + j];
}

// ---------------- host orchestration ----------------
extern "C" void kernel_launch(void* const* d_in, const int* in_sizes, int n_in,
                              void* d_out, int out_size, void* d_ws, size_t ws_size,
                              hipStream_t stream) {
    const float* l_feat = (const float*)d_in[0];
    const float* r_feat = (const float*)d_in[1];
    const float* Wall   = (const float*)d_in[2];
    const float* a_src  = (const float*)d_in[3];
    const float* a_dst  = (const float*)d_in[4];
    const float* bparam = (const float*)d_in[5];

    // workspace layout (floats)
    float* h  = (float*)d_ws;
    float* hn = h  + (size_t)NT * DD;
    float* z  = hn + (size_t)NT * DD;
    float* el = z  + (size_t)NT * DD;
    float* er = el + (size_t)NT * HH_;
    float* mst = er + (size_t)NT * HH_;
    float* sst = mst + (size_t)NL * HH_;
    unsigned char* mask = (unsigned char*)(sst + (size_t)NL * HH_);

    // 1) normalize rows into z (nl first half, nr second half), build mask
    k_normalize<<<NL, 256, 0, stream>>>(l_feat, z);
    k_normalize<<<NL, 256, 0, stream>>>(r_feat, z + (size_t)NL * DD);
    k_mask_wmma<<<dim3(NL / 32, NL / 16), 32, 0, stream>>>(z + (size_t)NL * DD, z, mask);

    // 2) h = concat(node_l, node_r)
    {
        int n = NT * DD;
        k_build_h<<<(n + 255) / 256, 256, 0, stream>>>(l_feat, r_feat, h);
    }

    // 3) bipartite GAT stack
    for (int i = 0; i < LL; ++i) {
        const float* Wl  = Wall   + (size_t)i * DD * DD;
        const float* asl = a_src  + (size_t)i * HH_ * DHH;
        const float* adl = a_dst  + (size_t)i * HH_ * DHH;
        const float* bl  = bparam + (size_t)i * DD;
        k_gemm_wmma<<<dim3(DD / 64, NT / 16), 32, 0, stream>>>(h, Wl, z, NT, DD, DD);
        k_elr<<<(NT * HH_ + 255) / 256, 256, 0, stream>>>(z, asl, adl, el, er);
        k_att_stats<<<(NL * HH_) / 8, 256, 0, stream>>>(el, er, mask, mst, sst);
        k_addbias<<<(NL * DD + 255) / 256, 256, 0, stream>>>(z, bl, hn, NL);
        k_att_aggregate<<<dim3(NL / 16, HH_), 32, 0, stream>>>(z, el, er, mask, mst, sst, bl, hn);
        if (i < LL - 1) k_elu<<<(NT * DD + 255) / 256, 256, 0, stream>>>(hn, NT * DD);
        float* t = h; h = hn; hn = t;
    }

    // 4) pooling GAT stack
    for (int i = 0; i < LL; ++i) {
        const float* Wl  = Wall   + (size_t)i * DD * DD;
        const float* asl = a_src  + (size_t)i * HH_ * DHH;
        const float* adl = a_dst  + (size_t)i * HH_ * DHH;
        const float* bl  = bparam + (size_t)i * DD;
        k_gemm_wmma<<<dim3(DD / 64, NT / 16), 32, 0, stream>>>(h, Wl, z, NT, DD, DD);
        k_elr<<<(NT * HH_ + 255) / 256, 256, 0, stream>>>(z, asl, adl, el, er);
        k_addbias<<<(NT * DD + 255) / 256, 256, 0, stream>>>(z, bl, hn, NT);
        k_pool<<<((NT / PP) * HH_ + 7) / 8, 256, 0, stream>>>(z, el, er, bl, hn);
        if (i < LL - 1) k_elu<<<(NT * DD + 255) / 256, 256, 0, stream>>>(hn, NT * DD);
        float* t = h; h = hn; hn = t;
    }

    // 5) gather image tokens
    k_gather<<<((NT / PP) * DD + 255) / 256, 256, 0, stream>>>(h, (float*)d_out);
}